// MHLDEPooling_51342039056683
// MI455X (gfx1250) — compile-verified
//
#include <hip/hip_runtime.h>
#include <hip/hip_bf16.h>

// Problem constants (match the reference)
#define BB 8
#define CC 1024
#define TT 400
#define HH 8
#define KK 64
#define DD 128   // CC / HH
#define NORM_EPS 1e-12f

typedef __attribute__((ext_vector_type(2))) float v2f;
typedef __attribute__((ext_vector_type(8))) float v8f;

// ---------------------------------------------------------------------------
// Kernel 1: per (b,h) compute z[k,t] = -scales[h,k] * (xx[t] - 2*dot + cc[k])
// dot = sum_d codes[h,d,k] * x[b,h,d,t]  via V_WMMA_F32_16X16X4_F32.
// Grid: B*H blocks of 128 threads (4 waves). Wave w owns k-tile [16w,16w+16).
// A fragments (codes^T) are invariant across the 25 t-tiles -> hoisted into
// registers once per wave (32 steps x v2f = 64 VGPRs).
// ---------------------------------------------------------------------------
__global__ __launch_bounds__(128) void k1_dist(const float* __restrict__ x,
                                               const float* __restrict__ codes,
                                               const float* __restrict__ scales,
                                               float* __restrict__ Z) {
  const int bh   = blockIdx.x;         // b*H + h
  const int b    = bh / HH;
  const int h    = bh % HH;
  const int tid  = threadIdx.x;
  const int lane = tid & 31;
  const int wave = tid >> 5;           // 0..3
  const int lh   = lane >> 4;          // lane-half: 0 or 1
  const int l16  = lane & 15;

  __shared__ float xx_lds[TT];         // sum_d x^2 for each t
  __shared__ float cc_lds[KK];         // sum_d codes^2 for each k
  __shared__ float sc_lds[KK];         // scales[h,k]

  const float* xbh = x + ((size_t)b * CC + (size_t)h * DD) * TT;  // [D][T]
  const float* ch  = codes + (size_t)h * DD * KK;                 // [D][K]

  for (int t = tid; t < TT; t += 128) {
    float s = 0.f;
    for (int d = 0; d < DD; ++d) { float v = xbh[(size_t)d * TT + t]; s += v * v; }
    xx_lds[t] = s;
  }
  for (int k = tid; k < KK; k += 128) {
    float s = 0.f;
    for (int d = 0; d < DD; ++d) { float v = ch[(size_t)d * KK + k]; s += v * v; }
    cc_lds[k] = s;
    sc_lds[k] = scales[h * KK + k];
  }
  __syncthreads();

  const int m0 = wave * 16;                       // k-code tile base

  // Hoist A (16x4 f32 fragments of codes^T): M = m0+l16, inner-K = 2*lh+vgpr.
  // Invariant over the whole nt loop.
  v2f afrag[DD / 4];
#pragma unroll
  for (int s = 0; s < DD / 4; ++s) {
    afrag[s].x = ch[(size_t)(4 * s + 2 * lh)     * KK + (m0 + l16)];
    afrag[s].y = ch[(size_t)(4 * s + 2 * lh + 1) * KK + (m0 + l16)];
  }

  float* Zbh = Z + (size_t)bh * KK * TT;

  for (int nt = 0; nt < TT / 16; ++nt) {          // 25 t-tiles
    const int t0 = nt * 16;
    v8f acc = {};
#pragma unroll
    for (int s = 0; s < DD / 4; ++s) {            // 32 WMMA steps over d
      // B (4x16 f32): N = t0+l16, inner-K = 2*lh + vgpr  -> x
      v2f bmat;
      bmat.x = xbh[(size_t)(4 * s + 2 * lh)     * TT + (t0 + l16)];
      bmat.y = xbh[(size_t)(4 * s + 2 * lh + 1) * TT + (t0 + l16)];
      acc = __builtin_amdgcn_wmma_f32_16x16x4_f32(
          false, afrag[s], false, bmat, (short)0, acc, false, false);
    }
    // C/D layout: element (M = m0 + v + 8*lh, N = t0 + l16)
    const int t = t0 + l16;
    const float xxv = xx_lds[t];
#pragma unroll
    for (int v = 0; v < 8; ++v) {
      const int k = m0 + v + 8 * lh;
      const float l2 = xxv - 2.f * acc[v] + cc_lds[k];
      Zbh[(size_t)k * TT + t] = -sc_lds[k] * l2;
    }
  }
}

// ---------------------------------------------------------------------------
// Kernel 2: softmax across heads (axis=1), numerically stable, in place.
// One thread per (b,k,t).
// ---------------------------------------------------------------------------
__global__ __launch_bounds__(256) void k2_softmax(float* __restrict__ Z) {
  const int idx   = blockIdx.x * 256 + threadIdx.x;
  const int total = BB * KK * TT;
  if (idx >= total) return;
  const int t = idx % TT;
  const int k = (idx / TT) % KK;
  const int b = idx / (KK * TT);
  const size_t base    = (size_t)b * HH * KK * TT + (size_t)k * TT + t;
  const size_t strideH = (size_t)KK * TT;

  float z[HH];
  float m = -3.402823466e38f;
  for (int h = 0; h < HH; ++h) {
    z[h] = Z[base + (size_t)h * strideH];
    m = fmaxf(m, z[h]);
  }
  float s = 0.f;
  for (int h = 0; h < HH; ++h) { z[h] = __expf(z[h] - m); s += z[h]; }
  const float inv = 1.f / s;
  for (int h = 0; h < HH; ++h) Z[base + (size_t)h * strideH] = z[h] * inv;
}

// ---------------------------------------------------------------------------
// Kernel 3: per (b,h):
//   E[d,k] = (sum_t x[d,t]*w[k,t] - codes[d,k]*sum_t w[k,t]) / T  (WMMA over t)
//   out[b,h,k,d] = E[d,k] / max(||E[:,k]||_2, eps)
// Grid: B*H blocks of 128 threads (4 waves). Wave handles 2 d-tiles; per
// d-tile it keeps 4 accumulators (one per k-tile) and streams T in chunks of
// 100 so each A fragment chunk (x, loaded as b64) is reused 4x.
// ---------------------------------------------------------------------------
__global__ __launch_bounds__(128) void k3_pool(const float* __restrict__ x,
                                               const float* __restrict__ codes,
                                               const float* __restrict__ W,
                                               float* __restrict__ out) {
  const int bh   = blockIdx.x;
  const int b    = bh / HH;
  const int h    = bh % HH;
  const int tid  = threadIdx.x;
  const int lane = tid & 31;
  const int wave = tid >> 5;
  const int lh   = lane >> 4;
  const int l16  = lane & 15;

  __shared__ float wsum_lds[KK];
  __shared__ float e_lds[KK * DD];   // [k][d], 32 KB
  __shared__ float inv_lds[KK];

  const float* xbh = x + ((size_t)b * CC + (size_t)h * DD) * TT;  // [D][T]
  const float* ch  = codes + (size_t)h * DD * KK;                 // [D][K]
  const float* Wbh = W + (size_t)bh * KK * TT;                    // [K][T]

  for (int k = tid; k < KK; k += 128) {
    float s = 0.f;
    for (int t = 0; t < TT; ++t) s += Wbh[(size_t)k * TT + t];
    wsum_lds[k] = s;
  }
  __syncthreads();

  const float invT = 1.0f / (float)TT;

  for (int mt = wave; mt < DD / 16; mt += 4) {    // 2 d-tiles per wave
    const int m0 = mt * 16;
    v8f acc[KK / 16] = {};                        // 4 accumulators (k-tiles)

    for (int tc = 0; tc < TT; tc += 100) {        // 4 chunks of 25 WMMA steps
      // A fragments for this chunk: M = m0+l16 (d), inner-K = tc+4s+2*lh+vgpr.
      // (t, t+1) contiguous -> single 8-byte load each.
      v2f afrag[25];
#pragma unroll
      for (int s = 0; s < 25; ++s)
        afrag[s] = *(const v2f*)(xbh + (size_t)(m0 + l16) * TT + tc + 4 * s + 2 * lh);

#pragma unroll
      for (int ntile = 0; ntile < KK / 16; ++ntile) {
        const int n0 = ntile * 16;
        v8f a = acc[ntile];
#pragma unroll
        for (int s = 0; s < 25; ++s) {
          // B: N = n0+l16 (k), inner-K = tc+4s+2*lh+vgpr -> w[k][t], (t,t+1)
          // contiguous -> b64 load.
          v2f bmat = *(const v2f*)(Wbh + (size_t)(n0 + l16) * TT + tc + 4 * s + 2 * lh);
          a = __builtin_amdgcn_wmma_f32_16x16x4_f32(
              false, afrag[s], false, bmat, (short)0, a, false, false);
        }
        acc[ntile] = a;
      }
    }

#pragma unroll
    for (int ntile = 0; ntile < KK / 16; ++ntile) {
      const int k = ntile * 16 + l16;
#pragma unroll
      for (int v = 0; v < 8; ++v) {
        const int d = m0 + v + 8 * lh;
        const float e = (acc[ntile][v] - ch[(size_t)d * KK + k] * wsum_lds[k]) * invT;
        e_lds[(size_t)k * DD + d] = e;
      }
    }
  }
  __syncthreads();

  if (tid < KK) {
    float s = 0.f;
    for (int d = 0; d < DD; ++d) { float e = e_lds[(size_t)tid * DD + d]; s += e * e; }
    inv_lds[tid] = 1.f / fmaxf(sqrtf(s), NORM_EPS);
  }
  __syncthreads();

  float* obh = out + (size_t)bh * KK * DD;       // out[b,h,k,d]
  for (int i = tid; i < KK * DD; i += 128) {
    const int k = i / DD;
    obh[i] = e_lds[i] * inv_lds[k];
  }
}

// ---------------------------------------------------------------------------
extern "C" void kernel_launch(void* const* d_in, const int* in_sizes, int n_in,
                              void* d_out, int out_size, void* d_ws, size_t ws_size,
                              hipStream_t stream) {
  const float* x      = (const float*)d_in[0];   // (B, C, T)
  const float* codes  = (const float*)d_in[1];   // (C, K)
  const float* scales = (const float*)d_in[2];   // (H, K)
  float* out = (float*)d_out;                    // (B, H, K, D)
  float* Z   = (float*)d_ws;                     // (B*H, K, T) logits -> weights

  k1_dist<<<BB * HH, 128, 0, stream>>>(x, codes, scales, Z);

  const int total = BB * KK * TT;
  k2_softmax<<<(total + 255) / 256, 256, 0, stream>>>(Z);

  k3_pool<<<BB * HH, 128, 0, stream>>>(x, codes, Z, out);
}